// Model_11888469476036
// MI455X (gfx1250) — compile-verified
//
#include <hip/hip_runtime.h>

typedef __attribute__((ext_vector_type(2))) float v2f;
typedef __attribute__((ext_vector_type(8))) float v8f;

#define EPSF 1e-5f

// Problem constants (fixed by the reference harness)
constexpr int B_  = 16, C_ = 32, D_ = 32, H_ = 64, W_ = 64;
constexpr int P_  = 8, HID = 256, OUT = 128;
constexpr int BIN4 = (D_ / P_) * H_ * W_ / 4;   // 4096 float4 per bin
constexpr int NBLK = C_ * B_ * P_;              // 4096 bins
constexpr float NCHAN = 2097152.0f;             // B*D*H*W elements per channel

// ---------------------------------------------------------------------------
// Kernel 1: single pass over x. Per (c,b,p) bin of 16384 contiguous floats:
// partial {sum, sumsq, max, min}. This is the only kernel that touches the
// 256 MiB tensor -> memory-bound at ~11.5us on 23.3 TB/s HBM.
// ---------------------------------------------------------------------------
__global__ void k1_stats_pool(const float* __restrict__ x,
                              float4* __restrict__ partials) {
    const int bid = blockIdx.x;        // bid = c*128 + b*8 + p
    const int c   = bid >> 7;
    const int rem = bid & 127;
    const int b   = rem >> 3;
    const int p   = rem & 7;
    const float4* src = (const float4*)(x +
        (((size_t)b * C_ + c) * D_ + (size_t)p * (D_ / P_)) * (size_t)(H_ * W_));

    const int t = threadIdx.x;
    float sum = 0.0f, ss = 0.0f;
    float mx = -__builtin_inff(), mn = __builtin_inff();
    for (int i = t; i < BIN4; i += 256) {
        float4 v = src[i];                               // global_load_b128
        sum += (v.x + v.y) + (v.z + v.w);
        ss  += (v.x * v.x + v.y * v.y) + (v.z * v.z + v.w * v.w);
        mx = fmaxf(mx, fmaxf(fmaxf(v.x, v.y), fmaxf(v.z, v.w)));
        mn = fminf(mn, fminf(fminf(v.x, v.y), fminf(v.z, v.w)));
    }
    __shared__ float s0[256], s1[256], s2[256], s3[256];
    s0[t] = sum; s1[t] = ss; s2[t] = mx; s3[t] = mn;
    __syncthreads();
    for (int off = 128; off > 0; off >>= 1) {
        if (t < off) {
            s0[t] += s0[t + off];
            s1[t] += s1[t + off];
            s2[t]  = fmaxf(s2[t], s2[t + off]);
            s3[t]  = fminf(s3[t], s3[t + off]);
        }
        __syncthreads();
    }
    if (t == 0) partials[bid] = make_float4(s0[0], s1[0], s2[0], s3[0]);
}

// ---------------------------------------------------------------------------
// Kernel 2: one block per channel. Reduce 128 partials -> mean/var, then
// pooled[p][b][c] = relu(a*extreme + bias) with extreme = (a>=0 ? max : min)
// (monotonicity of relu(affine) lets BN+ReLU+maxpool fuse into one x pass).
// ---------------------------------------------------------------------------
__global__ void k2_norm_pool(const float4* __restrict__ partials,
                             const float* __restrict__ bn_gamma,
                             const float* __restrict__ bn_beta,
                             float* __restrict__ pooled) {
    const int c = blockIdx.x;
    const int t = threadIdx.x;                 // 0..127 == b*8 + p
    float4 pv = partials[c * 128 + t];
    __shared__ float s0[128], s1[128];
    s0[t] = pv.x; s1[t] = pv.y;
    __syncthreads();
    for (int off = 64; off > 0; off >>= 1) {
        if (t < off) { s0[t] += s0[t + off]; s1[t] += s1[t + off]; }
        __syncthreads();
    }
    __shared__ float sh_a, sh_b;
    if (t == 0) {
        float mean = s0[0] / NCHAN;
        float var  = s1[0] / NCHAN - mean * mean;     // biased, matches ref
        float a    = bn_gamma[c] * rsqrtf(var + EPSF);
        sh_a = a;
        sh_b = bn_beta[c] - mean * a;
    }
    __syncthreads();
    const float a  = sh_a, bb = sh_b;
    const float ex = (a >= 0.0f) ? pv.z : pv.w;
    const float val = fmaxf(a * ex + bb, 0.0f);
    const int b = t >> 3, p = t & 7;
    pooled[(p * B_ + b) * C_ + c] = val;              // time-major (P,B,C)
}

// ---------------------------------------------------------------------------
// Kernel 3: RNN scan + projection + LayerNorm in one workgroup of 16 waves.
// Wave n owns output N-tile [16n,16n+16). GEMMs via V_WMMA_F32_16X16X4_F32.
// A layout: lanes 0-15 -> K=k0,k0+1 ; lanes 16-31 -> K=k0+2,k0+3 (row=lane%16)
// B layout: mirrored, N = lane%16. C/D: vgpr r -> row r + 8*(lane>=16).
// ---------------------------------------------------------------------------
__global__ void __launch_bounds__(512)
k3_rnn(const float* __restrict__ pooled,
       const float* __restrict__ w_ih, const float* __restrict__ b_ih,
       const float* __restrict__ w_hh, const float* __restrict__ b_hh,
       const float* __restrict__ w_out, const float* __restrict__ b_out,
       const float* __restrict__ ln_g, const float* __restrict__ ln_b,
       float* __restrict__ out) {
    __shared__ float hbuf[16 * HID];   // hidden state, 16 KB
    __shared__ float obuf[16 * OUT];   // projection result, 8 KB
    __shared__ float mu_s[16], rs_s[16];

    const int tid  = threadIdx.x;
    const int lane = tid & 31;
    const int wave = tid >> 5;         // 0..15
    const int col  = lane & 15;        // N (and A-row) index within tile
    const int half = lane >> 4;        // 0: K=k0..k0+1, 1: K=k0+2..k0+3
    const int j0   = wave * 16;

    for (int i = tid; i < 16 * HID; i += 512) hbuf[i] = 0.0f;
    __syncthreads();

    for (int step = 0; step < P_; ++step) {
        const float* xb = pooled + step * (B_ * C_);
        const float bias = b_ih[j0 + col] + b_hh[j0 + col];
        v8f acc;
        #pragma unroll
        for (int r = 0; r < 8; ++r) acc[r] = bias;

        // x_t @ W_ih^T  (K = 32)
        {
            const float* arow = xb + col * C_ + half * 2;
            const float* brow = w_ih + (size_t)(j0 + col) * C_ + half * 2;
            #pragma unroll
            for (int k0 = 0; k0 < C_; k0 += 4) {
                v2f av; av[0] = arow[k0]; av[1] = arow[k0 + 1];
                v2f bv; bv[0] = brow[k0]; bv[1] = brow[k0 + 1];
                acc = __builtin_amdgcn_wmma_f32_16x16x4_f32(
                          false, av, false, bv, (short)0, acc, false, false);
            }
        }
        // h @ W_hh^T  (K = 256)
        {
            const float* arow = hbuf + col * HID + half * 2;
            const float* brow = w_hh + (size_t)(j0 + col) * HID + half * 2;
            #pragma unroll 4
            for (int k0 = 0; k0 < HID; k0 += 4) {
                v2f av; av[0] = arow[k0]; av[1] = arow[k0 + 1];
                v2f bv; bv[0] = brow[k0]; bv[1] = brow[k0 + 1];
                acc = __builtin_amdgcn_wmma_f32_16x16x4_f32(
                          false, av, false, bv, (short)0, acc, false, false);
            }
        }
        __syncthreads();               // all readers of hbuf done
        #pragma unroll
        for (int r = 0; r < 8; ++r) {
            int m = r + half * 8;
            hbuf[m * HID + j0 + col] = fmaxf(acc[r], 0.0f);   // ReLU
        }
        __syncthreads();               // new h visible to everyone
    }

    // Output projection: h @ W_out^T + b_out  (OUT=128 -> waves 0..7)
    if (wave < 8) {
        v8f acc;
        const float biaso = b_out[j0 + col];
        #pragma unroll
        for (int r = 0; r < 8; ++r) acc[r] = biaso;
        const float* arow = hbuf + col * HID + half * 2;
        const float* brow = w_out + (size_t)(j0 + col) * HID + half * 2;
        #pragma unroll 4
        for (int k0 = 0; k0 < HID; k0 += 4) {
            v2f av; av[0] = arow[k0]; av[1] = arow[k0 + 1];
            v2f bv; bv[0] = brow[k0]; bv[1] = brow[k0 + 1];
            acc = __builtin_amdgcn_wmma_f32_16x16x4_f32(
                      false, av, false, bv, (short)0, acc, false, false);
        }
        #pragma unroll
        for (int r = 0; r < 8; ++r) {
            int m = r + half * 8;
            obuf[m * OUT + j0 + col] = acc[r];
        }
    }
    __syncthreads();

    // LayerNorm over OUT=128 per row
    if (tid < 16) {
        float s = 0.0f;
        for (int n = 0; n < OUT; ++n) s += obuf[tid * OUT + n];
        float mu = s / (float)OUT;
        float v = 0.0f;
        for (int n = 0; n < OUT; ++n) {
            float d = obuf[tid * OUT + n] - mu;
            v += d * d;
        }
        mu_s[tid] = mu;
        rs_s[tid] = rsqrtf(v / (float)OUT + EPSF);
    }
    __syncthreads();
    for (int i = tid; i < 16 * OUT; i += 512) {
        int m = i >> 7, n = i & 127;
        out[i] = (obuf[i] - mu_s[m]) * rs_s[m] * ln_g[n] + ln_b[n];
    }
}

// ---------------------------------------------------------------------------
extern "C" void kernel_launch(void* const* d_in, const int* in_sizes, int n_in,
                              void* d_out, int out_size, void* d_ws, size_t ws_size,
                              hipStream_t stream) {
    const float* x        = (const float*)d_in[0];
    const float* bn_gamma = (const float*)d_in[1];
    const float* bn_beta  = (const float*)d_in[2];
    const float* w_ih     = (const float*)d_in[3];
    const float* b_ih     = (const float*)d_in[4];
    const float* w_hh     = (const float*)d_in[5];
    const float* b_hh     = (const float*)d_in[6];
    const float* w_out    = (const float*)d_in[7];
    const float* b_out    = (const float*)d_in[8];
    const float* ln_g     = (const float*)d_in[9];
    const float* ln_b     = (const float*)d_in[10];
    float* out = (float*)d_out;

    float4* partials = (float4*)d_ws;                              // 64 KB
    float*  pooled   = (float*)((char*)d_ws + NBLK * sizeof(float4)); // 16 KB

    k1_stats_pool<<<NBLK, 256, 0, stream>>>(x, partials);
    k2_norm_pool<<<C_, 128, 0, stream>>>(partials, bn_gamma, bn_beta, pooled);
    k3_rnn<<<1, 512, 0, stream>>>(pooled, w_ih, b_ih, w_hh, b_hh,
                                  w_out, b_out, ln_g, ln_b, out);
}